// CondConv_Spatial_74139725463922
// MI455X (gfx1250) — compile-verified
//
#include <hip/hip_runtime.h>
#include <hip/hip_bf16.h>
#include <math.h>

typedef __attribute__((ext_vector_type(2))) float v2f;
typedef __attribute__((ext_vector_type(4))) float v4f;
typedef __attribute__((ext_vector_type(8))) float v8f;

// ---------------- problem constants ----------------
#define BS   8
#define CIN  64
#define HH   256
#define WW   256
#define NEXP 4
#define OCH  64
#define KS   3
#define KTOT (CIN * KS * KS)      // 576
#define WROW (CIN * KS * KS)      // per-o row length in agg_w, 576

// conv-kernel tiling
#define TILE_W   128              // output positions per block
#define CI_CHUNK 16               // input channels staged per LDS pass
#define KCHUNK   (CI_CHUNK * 9)   // 144 K values per pass
#define SW_STRIDE 148             // padded LDS stride for weight slab (bank-conflict free A reads)
#define PATCH_W  132              // 130 used (128 + 2 halo), padded

// im2col offset for K index k (chunk-local): k = ci*9 + kh*3 + kw
__device__ constexpr int poff(int k) {
    return ((k / 9) * 3 + ((k % 9) / 3)) * PATCH_W + (k % 3);
}

// ---------------- kernel 1: per-(b,c) row/col partial sums ----------------
__global__ __launch_bounds__(256)
void att_partial_kernel(const float* __restrict__ x,
                        float* __restrict__ parth,   // [BS][CIN][HH]
                        float* __restrict__ partv) { // [BS][CIN][WW]
    const int t = threadIdx.x;                       // 0..255
    const float* plane = x + (size_t)blockIdx.x * (HH * WW);
    float rowAcc = 0.f, colAcc = 0.f;
    for (int w = 0; w < WW; ++w) rowAcc += plane[t * WW + w];
    for (int h = 0; h < HH; ++h) colAcc += plane[h * WW + t];
    parth[(size_t)blockIdx.x * HH + t] = rowAcc;
    partv[(size_t)blockIdx.x * WW + t] = colAcc;
}

// ---------------- kernel 2: finalize attention + agg bias ----------------
__global__ __launch_bounds__(256)
void att_finalize_kernel(const float* __restrict__ parth,
                         const float* __restrict__ partv,
                         const float* __restrict__ net_w,   // [NEXP][HH+WW]
                         const float* __restrict__ bias,    // [NEXP][OCH]
                         float* __restrict__ att,           // [BS][NEXP]
                         float* __restrict__ aggb) {        // [BS][OCH]
    __shared__ float sh[HH + WW];
    __shared__ float sa[NEXP];
    const int b = blockIdx.x, t = threadIdx.x;
    float accH = 0.f, accV = 0.f;
    for (int c = 0; c < CIN; ++c) {
        accH += parth[((size_t)b * CIN + c) * HH + t];
        accV += partv[((size_t)b * CIN + c) * WW + t];
    }
    const float inv = 1.f / (float)(CIN * WW);
    sh[t]      = accH * inv;
    sh[HH + t] = accV * inv;
    __syncthreads();
    if (t < NEXP) {
        float acc = 0.f;
        for (int j = 0; j < HH + WW; ++j) acc += sh[j] * net_w[t * (HH + WW) + j];
        float s = 1.f / (1.f + __expf(-acc));
        att[b * NEXP + t] = s;
        sa[t] = s;
    }
    __syncthreads();
    if (t < OCH) {
        float acc = 0.f;
        for (int k = 0; k < NEXP; ++k) acc += sa[k] * bias[k * OCH + t];
        aggb[b * OCH + t] = acc;
    }
}

// ---------------- kernel 3: mix expert weights ----------------
__global__ __launch_bounds__(256)
void agg_weight_kernel(const float* __restrict__ weight,  // [NEXP][OCH*KTOT]
                       const float* __restrict__ att,     // [BS][NEXP]
                       float* __restrict__ aggw) {        // [BS][OCH*KTOT]
    const size_t idx = (size_t)blockIdx.x * 256 + threadIdx.x;
    const int b = (int)(idx / (OCH * KTOT));
    const int j = (int)(idx % (OCH * KTOT));
    const float a0 = att[b * NEXP + 0], a1 = att[b * NEXP + 1];
    const float a2 = att[b * NEXP + 2], a3 = att[b * NEXP + 3];
    const size_t E = (size_t)OCH * KTOT;
    aggw[idx] = a0 * weight[j] + a1 * weight[E + j] + a2 * weight[2 * E + j] + a3 * weight[3 * E + j];
}

// ---------------- kernel 4: per-sample 3x3 conv as implicit GEMM via WMMA f32 ----------------
// Block: (b, h, 128-wide w strip). M = 64 out channels, N = 128 positions, K = 576.
// 8 waves in a 2(M) x 4(N) grid; each wave owns a 32x32 register block = 2x2 WMMA tiles,
// reusing each A and B fragment twice per K-step.
__global__ __launch_bounds__(256)
void condconv_wmma_kernel(const float* __restrict__ x,
                          const float* __restrict__ aggw,
                          const float* __restrict__ aggb,
                          float* __restrict__ out) {
    __shared__ float sW[OCH * SW_STRIDE];          // 64 x 144 weight slab (padded stride)
    __shared__ float sX[CI_CHUNK * 3 * PATCH_W];   // 16ch x 3rows x 132cols input patch
    __shared__ float sBias[OCH];

    const int tid = threadIdx.x;
    const int b      = blockIdx.x >> 9;            // / (HH * (WW/TILE_W)) = /512
    const int rem    = blockIdx.x & 511;
    const int h      = rem >> 1;
    const int wstart = (rem & 1) * TILE_W;

    if (tid < OCH) sBias[tid] = aggb[b * OCH + tid];

    const int lane   = tid & 31;
    const int wv     = tid >> 5;
    const int hl     = lane >> 4;       // 0: K pair {0,1}; 1: K pair {2,3}
    const int l16    = lane & 15;
    const int m_base = (wv & 1) * 32;   // wave owns M rows [m_base, m_base+32)
    const int n_base = (wv >> 1) * 32;  // wave owns N cols [n_base, n_base+32)

    v8f acc[2][2] = {};

    const float* wrow = aggw + (size_t)b * (OCH * WROW);

    // per-thread LDS base addresses for the two A fragments (ds immediate adds kc0)
    const int aoff0 = (m_base + l16) * SW_STRIDE + 2 * hl;
    const int aoff1 = (m_base + 16 + l16) * SW_STRIDE + 2 * hl;
    const int n0 = n_base + l16;
    const int n1 = n0 + 16;

    for (int ci0 = 0; ci0 < CIN; ci0 += CI_CHUNK) {
        __syncthreads();
        // stage weight slab as float4: rows o, cols kc = [ci0*9, ci0*9+144)
        // src offset o*576 + ci0*9 + 4*kc4 and dst offset o*148 + 4*kc4 are both 16B aligned
        for (int e4 = tid; e4 < OCH * (KCHUNK / 4); e4 += 256) {   // 2304 float4s
            const int o = e4 / (KCHUNK / 4), kc4 = e4 % (KCHUNK / 4);
            const v4f v = *(const v4f*)&wrow[o * WROW + ci0 * 9 + kc4 * 4];
            *(v4f*)&sW[o * SW_STRIDE + kc4 * 4] = v;
        }
        // stage input patch with halo + zero padding at borders
        const float* xb = x + ((size_t)b * CIN + ci0) * (HH * WW);
        for (int e = tid; e < CI_CHUNK * 3 * PATCH_W; e += 256) {
            const int ci = e / (3 * PATCH_W);
            const int rr = (e / PATCH_W) % 3;
            const int cc = e % PATCH_W;
            const int gh = h - 1 + rr;
            const int gw = wstart - 1 + cc;
            float v = 0.f;
            if ((unsigned)gh < (unsigned)HH && (unsigned)gw < (unsigned)WW)
                v = xb[(size_t)ci * (HH * WW) + gh * WW + gw];
            sX[e] = v;
        }
        __syncthreads();

        // prefetch next chunk into cache while the XDL pipe works on this one
        if (ci0 + CI_CHUNK < CIN) {
            {   // next weight slab: 64 rows x 144 floats
                const int o = tid >> 2, seg = tid & 3;
                __builtin_prefetch(wrow + (size_t)o * WROW + (ci0 + CI_CHUNK) * 9 + seg * 36, 0, 1);
            }
            if (tid < 96) {  // next patch: 16ch x 3rows, two 64-float halves each
                const int ci = tid / 6, rr = (tid / 2) % 3, hs = tid & 1;
                int gh = h - 1 + rr; gh = gh < 0 ? 0 : (gh > HH - 1 ? HH - 1 : gh);
                int gw = wstart + hs * 64; if (wstart == 0 && hs == 0) gw = 0;
                const float* nx = x + ((size_t)b * CIN + ci0 + CI_CHUNK + ci) * (HH * WW);
                __builtin_prefetch(nx + gh * WW + gw, 0, 1);
            }
        }

        #pragma unroll
        for (int kc0 = 0; kc0 < KCHUNK; kc0 += 4) {
            // compile-time im2col offsets for k = kc0 + {0,1,2,3}; lane-half select
            const int o0 = hl ? poff(kc0 + 2) : poff(kc0 + 0);
            const int o1 = hl ? poff(kc0 + 3) : poff(kc0 + 1);
            // A fragments: 8B-aligned contiguous pair -> ds_load_b64
            const v2f a0 = *(const v2f*)&sW[aoff0 + kc0];
            const v2f a1 = *(const v2f*)&sW[aoff1 + kc0];
            // B fragments for the two N tiles
            v2f b0, b1;
            b0.x = sX[o0 + n0]; b0.y = sX[o1 + n0];
            b1.x = sX[o0 + n1]; b1.y = sX[o1 + n1];
            acc[0][0] = __builtin_amdgcn_wmma_f32_16x16x4_f32(false, a0, false, b0, (short)0, acc[0][0], false, false);
            acc[0][1] = __builtin_amdgcn_wmma_f32_16x16x4_f32(false, a0, false, b1, (short)0, acc[0][1], false, false);
            acc[1][0] = __builtin_amdgcn_wmma_f32_16x16x4_f32(false, a1, false, b0, (short)0, acc[1][0], false, false);
            acc[1][1] = __builtin_amdgcn_wmma_f32_16x16x4_f32(false, a1, false, b1, (short)0, acc[1][1], false, false);
        }
    }

    // epilogue: C/D layout — VGPR r: M=r (lanes 0-15) / M=r+8 (lanes 16-31), N=l16
    #pragma unroll
    for (int mi = 0; mi < 2; ++mi) {
        #pragma unroll
        for (int nj = 0; nj < 2; ++nj) {
            const int n = n_base + nj * 16 + l16;
            #pragma unroll
            for (int r = 0; r < 8; ++r) {
                const int o = m_base + mi * 16 + hl * 8 + r;
                out[(((size_t)b * OCH + o) * HH + h) * WW + wstart + n] = acc[mi][nj][r] + sBias[o];
            }
        }
    }
}

// ---------------- launch ----------------
extern "C" void kernel_launch(void* const* d_in, const int* in_sizes, int n_in,
                              void* d_out, int out_size, void* d_ws, size_t ws_size,
                              hipStream_t stream) {
    const float* x      = (const float*)d_in[0];   // (8,64,256,256)
    const float* net_w  = (const float*)d_in[1];   // (4,512)
    const float* weight = (const float*)d_in[2];   // (4,64,64,3,3)
    const float* bias   = (const float*)d_in[3];   // (4,64)
    float* out = (float*)d_out;

    // workspace layout (floats)
    float* ws = (float*)d_ws;
    float* att   = ws;                                   // 32
    float* aggb  = att + BS * NEXP;                      // 512
    float* aggw  = aggb + BS * OCH;                      // 294912
    float* parth = aggw + (size_t)BS * OCH * KTOT;       // 131072
    float* partv = parth + (size_t)BS * CIN * HH;        // 131072

    att_partial_kernel<<<BS * CIN, 256, 0, stream>>>(x, parth, partv);
    att_finalize_kernel<<<BS, 256, 0, stream>>>(parth, partv, net_w, bias, att, aggb);
    agg_weight_kernel<<<(BS * OCH * KTOT) / 256, 256, 0, stream>>>(weight, att, aggw);
    condconv_wmma_kernel<<<BS * HH * (WW / TILE_W), 256, 0, stream>>>(x, aggw, aggb, out);
}